// TtLlamaAttention_galaxy_51634096833072
// MI455X (gfx1250) — compile-verified
//
#include <hip/hip_runtime.h>

typedef __attribute__((ext_vector_type(2))) float v2f;
typedef __attribute__((ext_vector_type(4))) float f4;
typedef __attribute__((ext_vector_type(8))) float v8f;

#define HIDDEN 8192
#define NHEADS 64
#define NKV    8
#define HD     128
#define BATCH  32
#define PADLEN 2048

// ---------------------------------------------------------------------------
// fp32 GEMM: C[M,N] (+)= A[M,K_chunk] * W[K_chunk,N] via V_WMMA_F32_16X16X4_F32.
// K and N are compile-time so all W/A addressing strength-reduces to
// immediate-offset loads (no s_mul_u64 in the hot loop). KSPLIT>1 partitions K
// across blockIdx.z for occupancy; partial tiles are combined with
// global_atomic_add_f32 into a zeroed C. MT stacked 16x16 M-tiles share B.
// A operand (16x4 f32): lane lr holds row m0+lr, K pair (k,k+1) lo-half /
// (k+2,k+3) hi-half -> one contiguous b64 per lane.
// B operand (4x16 f32): VGPR0/1 = rows k,k+1 (lo) / k+2,k+3 (hi), col n0+lr.
// ---------------------------------------------------------------------------
template <int MT, int KSPLIT, int K, int N>
__global__ __launch_bounds__(256) void gemm_f32_wmma(
    const float* __restrict__ A, const float* __restrict__ W,
    float* __restrict__ C)
{
  constexpr int KC = K / KSPLIT;
  const int wave = threadIdx.x >> 5;
  const int lane = threadIdx.x & 31;
  const int lr   = lane & 15;
  const int hi   = lane >> 4;
  const int n0   = (blockIdx.x * 8 + wave) * 16;
  const int m0   = blockIdx.y * (16 * MT);
  const int k0   = blockIdx.z * KC;

  const float* Wp = W + (size_t)(k0 + 2 * hi) * N + n0 + lr;
  const float* Ap[MT];
#pragma unroll
  for (int t = 0; t < MT; ++t)
    Ap[t] = A + (size_t)(m0 + 16 * t + lr) * K + k0 + 2 * hi;

  v8f acc[MT] = {};

#pragma unroll 4
  for (int k = 0; k < KC; k += 4) {
    v2f bb;
    bb.x = __builtin_nontemporal_load(&Wp[(size_t)k * N]);
    bb.y = __builtin_nontemporal_load(&Wp[(size_t)(k + 1) * N]);
#pragma unroll
    for (int t = 0; t < MT; ++t) {
      v2f aa;
      aa.x = Ap[t][k];
      aa.y = Ap[t][k + 1];
      acc[t] = __builtin_amdgcn_wmma_f32_16x16x4_f32(
          false, aa, false, bb, (short)0, acc[t], false, false);
    }
    if ((k & 15) == 0 && k + 256 < KC)   // prefetch weight stream 256 rows ahead
      __builtin_prefetch(&Wp[(size_t)(k + 256) * N], 0, 1);
  }

  // C/D layout: VGPR r -> row m0+16t+r (lanes 0-15) / +8+r (lanes 16-31)
#pragma unroll
  for (int t = 0; t < MT; ++t) {
    float* Cp = C + (size_t)(m0 + 16 * t + 8 * hi) * N + n0 + lr;
#pragma unroll
    for (int r = 0; r < 8; ++r) {
      if (KSPLIT == 1) Cp[(size_t)r * N] = acc[t][r];
      else             atomicAdd(&Cp[(size_t)r * N], acc[t][r]);
    }
  }
}

// ---------------------------------------------------------------------------
// Flash-decode attention, one WG per (batch, kv-head). Each wave processes
// TWO positions per iteration: lanes split as 2 halves x 16 dim-slices
// (8 dims = 32B per lane, rows stay fully-coalesced b128 NT loads). The
// score reduction is a 4-step intra-half butterfly serving both positions at
// once; each half-wave keeps an independent online-softmax accumulator.
// 16 partials merge through LDS (64KB of the 320KB WGP pool).
// KV-cache inputs are never mutated: the freshly rotated k / new v row is
// substituted at pos == start_pos.
// ---------------------------------------------------------------------------
__global__ __launch_bounds__(256) void attn_decode(
    const float* __restrict__ q_rot,   // (B, 64, 128)
    const float* __restrict__ k_rot,   // (B, 8, 128)
    const float* __restrict__ v_new,   // (B, 8, 128)
    const float* __restrict__ k_cache, // (B, 8, 2048, 128)
    const float* __restrict__ v_cache, // (B, 8, 2048, 128)
    const int* __restrict__ start_pos_p,
    float* __restrict__ attn_out)      // (B, 64, 128)
{
  const int kv   = blockIdx.x;   // 0..7
  const int b    = blockIdx.y;   // 0..31
  const int wave = threadIdx.x >> 5;
  const int lane = threadIdx.x & 31;
  const int half = lane >> 4;          // which of the 2 positions this iter
  const int sl   = lane & 15;          // dim-slice within the half
  const int d0   = sl * 8;             // dims [d0, d0+8)
  const int part = wave * 2 + half;    // 16 independent accumulators
  const int sp   = start_pos_p[0];
  const float scale = 0.088388347648318447f; // 1/sqrt(128)

  __shared__ float lm[16][8];
  __shared__ float ls[16][8];
  __shared__ float lo[16][8][HD];      // 64 KB

  const float* qb = q_rot + (size_t)(b * NHEADS + kv * 8) * HD;
  f4 qv[8][2];
#pragma unroll
  for (int g = 0; g < 8; ++g) {
    qv[g][0] = *(const f4*)(qb + g * HD + d0);
    qv[g][1] = *(const f4*)(qb + g * HD + d0 + 4);
  }

  const float* Kb = k_cache + (size_t)(b * NKV + kv) * PADLEN * HD;
  const float* Vb = v_cache + (size_t)(b * NKV + kv) * PADLEN * HD;
  const float* kn = k_rot + (size_t)(b * NKV + kv) * HD;
  const float* vn = v_new + (size_t)(b * NKV + kv) * HD;

  float m[8], s[8];
  f4 o[8][2];
#pragma unroll
  for (int g = 0; g < 8; ++g) {
    m[g] = -3.0e38f; s[g] = 0.0f;
    o[g][0] = (f4){0.f, 0.f, 0.f, 0.f};
    o[g][1] = (f4){0.f, 0.f, 0.f, 0.f};
  }

  for (int base = wave * 2; base <= sp; base += 16) {
    const int pos = base + half;
    if (pos <= sp) {                       // per-half; shuffles stay intra-half
      f4 k40, k41, v40, v41;
      if (pos == sp) {
        k40 = *(const f4*)(kn + d0); k41 = *(const f4*)(kn + d0 + 4);
        v40 = *(const f4*)(vn + d0); v41 = *(const f4*)(vn + d0 + 4);
      } else {
        k40 = __builtin_nontemporal_load((const f4*)(Kb + (size_t)pos * HD + d0));
        k41 = __builtin_nontemporal_load((const f4*)(Kb + (size_t)pos * HD + d0 + 4));
        v40 = __builtin_nontemporal_load((const f4*)(Vb + (size_t)pos * HD + d0));
        v41 = __builtin_nontemporal_load((const f4*)(Vb + (size_t)pos * HD + d0 + 4));
      }
      float sc[8];
#pragma unroll
      for (int g = 0; g < 8; ++g) {
        f4 t = qv[g][0] * k40 + qv[g][1] * k41;
        float p = t.x + t.y + t.z + t.w;
#pragma unroll
        for (int off = 8; off > 0; off >>= 1)   // 4-step intra-half butterfly
          p += __shfl_xor(p, off, 32);
        sc[g] = p * scale;
      }
#pragma unroll
      for (int g = 0; g < 8; ++g) {
        float nm   = fmaxf(m[g], sc[g]);
        float corr = __expf(m[g] - nm);
        float p    = __expf(sc[g] - nm);
        s[g]    = s[g] * corr + p;
        o[g][0] = o[g][0] * corr + v40 * p;
        o[g][1] = o[g][1] * corr + v41 * p;
        m[g]    = nm;
      }
    }
  }

  if (sl == 0) {
#pragma unroll
    for (int g = 0; g < 8; ++g) { lm[part][g] = m[g]; ls[part][g] = s[g]; }
  }
#pragma unroll
  for (int g = 0; g < 8; ++g) {
    *(f4*)(&lo[part][g][d0])     = o[g][0];
    *(f4*)(&lo[part][g][d0 + 4]) = o[g][1];
  }
  __syncthreads();

  // wave g merges head g across the 16 half-wave partials
  const int g = wave;
  float M = -3.0e38f;
#pragma unroll
  for (int w = 0; w < 16; ++w) M = fmaxf(M, lm[w][g]);
  float S = 0.0f;
  f4 acc = (f4){0.f, 0.f, 0.f, 0.f};
#pragma unroll
  for (int w = 0; w < 16; ++w) {
    float f = __expf(lm[w][g] - M);
    S += ls[w][g] * f;
    acc += (*(const f4*)(&lo[w][g][4 * lane])) * f;
  }
  f4 res = acc * (1.0f / S);
  *(f4*)(attn_out + (size_t)(b * NHEADS + kv * 8 + g) * HD + 4 * lane) = res;
}

// ---------------------------------------------------------------------------
extern "C" void kernel_launch(void* const* d_in, const int* in_sizes, int n_in,
                              void* d_out, int out_size, void* d_ws, size_t ws_size,
                              hipStream_t stream) {
  const float* x   = (const float*)d_in[0];
  const float* wq  = (const float*)d_in[1];
  const float* wk  = (const float*)d_in[2];
  const float* wv  = (const float*)d_in[3];
  const float* wo  = (const float*)d_in[4];
  const float* rot = (const float*)d_in[5];
  const float* kc  = (const float*)d_in[6];
  const float* vc  = (const float*)d_in[7];
  const int*   sp  = (const int*)d_in[8];
  float*       out = (float*)d_out;

  // workspace layout (fp32): q,kx,vx are atomic-accumulated -> zeroed below
  float* q  = (float*)d_ws;                 // (32, 8192)
  float* kx = q  + BATCH * NHEADS * HD;     // (32, 1024)
  float* vx = kx + BATCH * NKV * HD;        // (32, 1024)
  float* qr = vx + BATCH * NKV * HD;        // (32, 8192)
  float* kr = qr + BATCH * NHEADS * HD;     // (32, 1024)
  float* ao = kr + BATCH * NKV * HD;        // (32, 8192)

  // zero the K-split accumulation targets (graph-capturable memsets)
  hipMemsetAsync(q,   0, (size_t)BATCH * (NHEADS + 2 * NKV) * HD * 4, stream);
  hipMemsetAsync(out, 0, (size_t)BATCH * HIDDEN * 4, stream);

  dim3 blk(256);

  // q/k/v projections: K split 8 ways for occupancy (4096 / 512 waves)
  gemm_f32_wmma<2, 8, HIDDEN, NHEADS * HD>
      <<<dim3((NHEADS * HD) / 128, 1, 8), blk, 0, stream>>>(x, wq, q);
  gemm_f32_wmma<2, 8, HIDDEN, NKV * HD>
      <<<dim3((NKV * HD) / 128, 1, 8), blk, 0, stream>>>(x, wk, kx);
  gemm_f32_wmma<2, 8, HIDDEN, NKV * HD>
      <<<dim3((NKV * HD) / 128, 1, 8), blk, 0, stream>>>(x, wv, vx);

  // rotary embedding as (rows,128)x(128,128) GEMMs
  gemm_f32_wmma<1, 1, HD, HD>
      <<<dim3(1, (BATCH * NHEADS) / 16, 1), blk, 0, stream>>>(q, rot, qr);
  gemm_f32_wmma<1, 1, HD, HD>
      <<<dim3(1, (BATCH * NKV) / 16, 1), blk, 0, stream>>>(kx, rot, kr);

  // flash-decode attention over the KV cache (+ in-flight new k/v row)
  attn_decode<<<dim3(NKV, BATCH), blk, 0, stream>>>(qr, kr, vx, kc, vc, sp, ao);

  // output projection (atomic K-split into zeroed d_out)
  gemm_f32_wmma<2, 8, HIDDEN, HIDDEN>
      <<<dim3(HIDDEN / 128, 1, 8), blk, 0, stream>>>(ao, wo, out);
}